// HeteroSAGE_67362267070926
// MI455X (gfx1250) — compile-verified
//
#include <hip/hip_runtime.h>
#include <hip/hip_bf16.h>

typedef __attribute__((ext_vector_type(2))) float v2f;
typedef __attribute__((ext_vector_type(8))) float v8f;

#define D 128
#define TILE_ROWS 32
#define GEMM_THREADS 512   // 16 waves: 2 row-tiles x 8 col-tiles of 16x16

// ---------------------------------------------------------------------------
// degree: deg[dst[e]] += 1
// ---------------------------------------------------------------------------
__global__ void degree_kernel(const int* __restrict__ dst, float* __restrict__ deg, int E) {
    int e = blockIdx.x * blockDim.x + threadIdx.x;
    if (e < E) atomicAdd(&deg[dst[e]], 1.0f);
}

// deg[i] = 1 / max(deg[i], 1)
__global__ void invdeg_kernel(float* __restrict__ deg, int N) {
    int i = blockIdx.x * blockDim.x + threadIdx.x;
    if (i < N) deg[i] = 1.0f / fmaxf(deg[i], 1.0f);
}

// ---------------------------------------------------------------------------
// scatter-add of gathered rows: one wave per edge, 4 floats per lane.
// agg (25.6MB) is L2-resident -> atomics resolve at L2 throughput.
// ---------------------------------------------------------------------------
__global__ void scatter_kernel(const float* __restrict__ xsrc,
                               const int* __restrict__ src,
                               const int* __restrict__ dst,
                               float* __restrict__ agg, int E) {
    int e    = blockIdx.x * 8 + (threadIdx.x >> 5);
    int lane = threadIdx.x & 31;
    if (e >= E) return;
    int s = src[e];
    int d = dst[e];
    const float4 v = *(const float4*)(xsrc + (size_t)s * D + lane * 4);
    float* ap = agg + (size_t)d * D + lane * 4;
    atomicAdd(ap + 0, v.x);
    atomicAdd(ap + 1, v.y);
    atomicAdd(ap + 2, v.z);
    atomicAdd(ap + 3, v.w);
}

// ---------------------------------------------------------------------------
// Fused SAGE GEMM:
//   out[m,:] = relu?( (invdeg[m]*agg[m,:]) @ Wl  +  root[m,:] @ Wr  +  bias )
// One workgroup = 32 rows x 128 cols. 16 waves, each owns a 16x16 tile.
// V_WMMA_F32_16X16X4_F32, K=4 steps, 64 WMMAs per wave total.
// ---------------------------------------------------------------------------
#define LDS_STRIDE 132   // 128 + 4 pad: bank stride 4 -> conflict-free b64 reads

__device__ __forceinline__ void stage_tile(float (*lds)[LDS_STRIDE],
                                           const float* __restrict__ X,
                                           const float* __restrict__ rowscale,
                                           int m0, int N, int tid) {
    // 32 rows x 128 floats = 1024 float4 slots; 512 threads x 2 iterations.
#pragma unroll
    for (int it = 0; it < 2; ++it) {
        int idx = tid + it * GEMM_THREADS;
        int r   = idx >> 5;          // 0..31
        int c4  = (idx & 31) << 2;   // 0,4,...,124
        int row = m0 + r;
        float4 v = make_float4(0.f, 0.f, 0.f, 0.f);
        if (row < N) {
            v = *(const float4*)(X + (size_t)row * D + c4);
            if (rowscale) {
                float s = rowscale[row];
                v.x *= s; v.y *= s; v.z *= s; v.w *= s;
            }
        }
        *(float4*)&lds[r][c4] = v;
    }
}

__device__ __forceinline__ v8f kloop_wmma(v8f c, const float (*lds)[LDS_STRIDE],
                                          const float* __restrict__ W,
                                          int mB, int ml, int hi, int n0) {
#pragma unroll
    for (int k0 = 0; k0 < D; k0 += 4) {
        int kb = k0 + 2 * hi;                 // lanes 0-15: K={k0,k0+1}; 16-31: K={k0+2,k0+3}
        v2f a = *(const v2f*)&lds[mB + ml][kb];
        v2f b;
        b.x = W[(size_t)kb * D + n0 + ml];
        b.y = W[(size_t)(kb + 1) * D + n0 + ml];
        c = __builtin_amdgcn_wmma_f32_16x16x4_f32(
                /*neg_a=*/false, a, /*neg_b=*/false, b,
                /*c_mod=*/(short)0, c, /*reuse_a=*/false, /*reuse_b=*/false);
    }
    return c;
}

__global__ void __launch_bounds__(GEMM_THREADS)
sage_gemm_kernel(const float* __restrict__ agg, const float* __restrict__ invdeg,
                 const float* __restrict__ Wl,
                 const float* __restrict__ root, const float* __restrict__ Wr,
                 const float* __restrict__ bias,
                 float* __restrict__ out, int N, int do_relu) {
    __shared__ float ldsA[TILE_ROWS][LDS_STRIDE];

    const int tid  = threadIdx.x;
    const int m0   = blockIdx.x * TILE_ROWS;
    const int wave = tid >> 5;
    const int lane = tid & 31;
    const int hi   = lane >> 4;       // half-wave select
    const int ml   = lane & 15;
    const int mB   = (wave >> 3) << 4;  // 0 or 16 : row tile inside block
    const int n0   = (wave & 7) << 4;   // 0..112  : col tile

    // Warm the weight matrices into cache (global_prefetch_b8).
    if (tid < D) {
        __builtin_prefetch(Wl + (size_t)tid * D, 0, 0);
        __builtin_prefetch(Wr + (size_t)tid * D, 0, 0);
    }

    v8f c = {};

    // Pass 1: (invdeg * agg) @ Wl
    stage_tile(ldsA, agg, invdeg, m0, N, tid);
    __syncthreads();
    c = kloop_wmma(c, ldsA, Wl, mB, ml, hi, n0);
    __syncthreads();

    // Pass 2: root @ Wr into the same accumulator
    stage_tile(ldsA, root, nullptr, m0, N, tid);
    __syncthreads();
    c = kloop_wmma(c, ldsA, Wr, mB, ml, hi, n0);

    // Epilogue: bias (+ReLU), C layout: vgpr i -> M = i + 8*hi, N = ml
    const float bv = bias[n0 + ml];
#pragma unroll
    for (int i = 0; i < 8; ++i) {
        int row = m0 + mB + i + 8 * hi;
        if (row < N) {
            float v = c[i] + bv;
            if (do_relu) v = fmaxf(v, 0.f);
            out[(size_t)row * D + n0 + ml] = v;
        }
    }
}

// ---------------------------------------------------------------------------
// Host-side orchestration
// ---------------------------------------------------------------------------
extern "C" void kernel_launch(void* const* d_in, const int* in_sizes, int n_in,
                              void* d_out, int out_size, void* d_ws, size_t ws_size,
                              hipStream_t stream) {
    const float* x_user = (const float*)d_in[0];
    const float* x_item = (const float*)d_in[1];
    const int*   e_ui   = (const int*)d_in[2];
    const int*   e_iu   = (const int*)d_in[3];
    const float* W1l_ui = (const float*)d_in[4];
    const float* W1r_ui = (const float*)d_in[5];
    const float* W1l_iu = (const float*)d_in[6];
    const float* W1r_iu = (const float*)d_in[7];
    const float* W2l_ui = (const float*)d_in[8];
    const float* W2r_ui = (const float*)d_in[9];
    const float* W2l_iu = (const float*)d_in[10];
    const float* W2r_iu = (const float*)d_in[11];
    const float* b1_ui  = (const float*)d_in[12];
    const float* b1_iu  = (const float*)d_in[13];
    const float* b2_ui  = (const float*)d_in[14];
    const float* b2_iu  = (const float*)d_in[15];

    const int NU = in_sizes[0] / D;
    const int NI = in_sizes[1] / D;
    const int E  = in_sizes[2] / 2;

    const int* src_ui = e_ui;          // user indices
    const int* dst_ui = e_ui + E;      // item indices
    const int* src_iu = e_iu;          // item indices
    const int* dst_iu = e_iu + E;      // user indices

    const size_t nf_u = (size_t)NU * D;
    const size_t nf_i = (size_t)NI * D;

    float* ws    = (float*)d_ws;
    float* agg_i = ws;                 // [NI, D]
    float* agg_u = agg_i + nf_i;       // [NU, D]
    float* h1_i  = agg_u + nf_u;       // [NI, D]
    float* h1_u  = h1_i + nf_i;        // [NU, D]
    float* deg_i = h1_u + nf_u;        // [NI]
    float* deg_u = deg_i + NI;         // [NU]

    float* out_user = (float*)d_out;            // h2_user first (tuple order)
    float* out_item = out_user + nf_u;          // then h2_item

    const int deg_blocks = (E + 255) / 256;
    const int sc_blocks  = (E + 7) / 8;
    const int gi_blocks  = (NI + TILE_ROWS - 1) / TILE_ROWS;
    const int gu_blocks  = (NU + TILE_ROWS - 1) / TILE_ROWS;

    // ---- degrees (reused by both layers) ----
    hipMemsetAsync(deg_i, 0, (size_t)(NI + NU) * sizeof(float), stream);
    degree_kernel<<<deg_blocks, 256, 0, stream>>>(dst_ui, deg_i, E);
    degree_kernel<<<deg_blocks, 256, 0, stream>>>(dst_iu, deg_u, E);
    invdeg_kernel<<<(NI + 255) / 256, 256, 0, stream>>>(deg_i, NI);
    invdeg_kernel<<<(NU + 255) / 256, 256, 0, stream>>>(deg_u, NU);

    // ---- layer 1 aggregation ----
    hipMemsetAsync(agg_i, 0, (nf_i + nf_u) * sizeof(float), stream);
    scatter_kernel<<<sc_blocks, 256, 0, stream>>>(x_user, src_ui, dst_ui, agg_i, E);
    scatter_kernel<<<sc_blocks, 256, 0, stream>>>(x_item, src_iu, dst_iu, agg_u, E);

    // ---- layer 1 GEMMs (+ReLU) ----
    sage_gemm_kernel<<<gi_blocks, GEMM_THREADS, 0, stream>>>(
        agg_i, deg_i, W1l_ui, x_item, W1r_ui, b1_ui, h1_i, NI, 1);
    sage_gemm_kernel<<<gu_blocks, GEMM_THREADS, 0, stream>>>(
        agg_u, deg_u, W1l_iu, x_user, W1r_iu, b1_iu, h1_u, NU, 1);

    // ---- layer 2 aggregation (reuse accumulators) ----
    hipMemsetAsync(agg_i, 0, (nf_i + nf_u) * sizeof(float), stream);
    scatter_kernel<<<sc_blocks, 256, 0, stream>>>(h1_u, src_ui, dst_ui, agg_i, E);
    scatter_kernel<<<sc_blocks, 256, 0, stream>>>(h1_i, src_iu, dst_iu, agg_u, E);

    // ---- layer 2 GEMMs (no ReLU) -> outputs ----
    sage_gemm_kernel<<<gu_blocks, GEMM_THREADS, 0, stream>>>(
        agg_u, deg_u, W2l_iu, h1_u, W2r_iu, b2_iu, out_user, NU, 0);
    sage_gemm_kernel<<<gi_blocks, GEMM_THREADS, 0, stream>>>(
        agg_i, deg_i, W2l_ui, h1_i, W2r_ui, b2_ui, out_item, NI, 0);
}